// GATModel_54168127537295
// MI455X (gfx1250) — compile-verified
//
#include <hip/hip_runtime.h>
#include <math.h>

#define D_IN   128
#define HEADS  8
#define HID    64
#define F1     (HEADS * HID)   // 512
#define D_OUT  8
#define NEG_SLOPE 0.2f

typedef float v2f __attribute__((ext_vector_type(2)));
typedef float v8f __attribute__((ext_vector_type(8)));

// ---- float <-> ordered-uint encoding for atomic max on floats ----
__device__ __forceinline__ unsigned f2ord(float f) {
  unsigned u = __float_as_uint(f);
  return (u & 0x80000000u) ? ~u : (u | 0x80000000u);
}
__device__ __forceinline__ float ord2f(unsigned v) {
  unsigned u = (v & 0x80000000u) ? (v ^ 0x80000000u) : ~v;
  return __uint_as_float(u);
}
__device__ __forceinline__ float lrelu(float x) { return x > 0.f ? x : NEG_SLOPE * x; }

// =====================================================================
// GEMM1: h1[N,512] = x[N,128] @ W1[128,512]   (fp32 WMMA 16x16x4)
// one wave per 16x16 output tile; K-loop of 32 WMMA ops
// =====================================================================
__global__ void gemm1_wmma(const float* __restrict__ X, const float* __restrict__ W,
                           float* __restrict__ H, int n) {
  int wave = (blockIdx.x * blockDim.x + threadIdx.x) >> 5;
  int lane = threadIdx.x & 31;
  const int ntiles = F1 / 16;              // 32 column tiles
  int mt = wave / ntiles;
  int nt = wave - mt * ntiles;
  int row0 = mt * 16;
  int col0 = nt * 16;
  if (row0 >= n) return;                   // wave-uniform: EXEC stays all-ones

  int half = lane >> 4;                    // 0: K={k,k+1}, 1: K={k+2,k+3}
  int l15  = lane & 15;
  const float* xrow = X + (size_t)(row0 + l15) * D_IN;
  int bcol = col0 + l15;

  v8f acc = {};
  for (int k = 0; k < D_IN; k += 4) {
    int kk = k + half * 2;
    v2f a; a.x = xrow[kk];                       a.y = xrow[kk + 1];
    v2f b; b.x = W[(size_t)kk * F1 + bcol];      b.y = W[(size_t)(kk + 1) * F1 + bcol];
    acc = __builtin_amdgcn_wmma_f32_16x16x4_f32(false, a, false, b,
                                                (short)0, acc, false, false);
  }
#pragma unroll
  for (int v = 0; v < 8; ++v) {
    int r = row0 + v + half * 8;           // C/D layout: vgpr v, half-wave -> M
    H[(size_t)r * F1 + bcol] = acc[v];
  }
}

// =====================================================================
// GEMM2: h2[N,8] = x2[N,512] @ W2[512,8]  (WMMA, cols 8..15 padded 0)
// =====================================================================
__global__ void gemm2_wmma(const float* __restrict__ X2, const float* __restrict__ W2,
                           float* __restrict__ H2, int n) {
  int wave = (blockIdx.x * blockDim.x + threadIdx.x) >> 5;
  int lane = threadIdx.x & 31;
  int row0 = wave * 16;
  if (row0 >= n) return;                   // wave-uniform

  int half = lane >> 4;
  int l15  = lane & 15;
  const float* xrow = X2 + (size_t)(row0 + l15) * F1;
  bool colok = (l15 < D_OUT);

  v8f acc = {};
  for (int k = 0; k < F1; k += 4) {
    int kk = k + half * 2;
    v2f a; a.x = xrow[kk]; a.y = xrow[kk + 1];
    v2f b;
    b.x = colok ? W2[(size_t)kk * D_OUT + l15]       : 0.f;  // select, no branch
    b.y = colok ? W2[(size_t)(kk + 1) * D_OUT + l15] : 0.f;
    acc = __builtin_amdgcn_wmma_f32_16x16x4_f32(false, a, false, b,
                                                (short)0, acc, false, false);
  }
  if (colok) {
#pragma unroll
    for (int v = 0; v < 8; ++v) {
      int r = row0 + v + half * 8;
      H2[(size_t)r * D_OUT + l15] = acc[v];
    }
  }
}

// =====================================================================
// attention coefficients layer 1: as/ad[n,h] = sum_c h1[n,h,c]*a[h,c]
// =====================================================================
__global__ void alphas1(const float* __restrict__ H, const float* __restrict__ a_src,
                        const float* __restrict__ a_dst,
                        float* __restrict__ as, float* __restrict__ ad, int n) {
  int t = blockIdx.x * blockDim.x + threadIdx.x;
  if (t >= n * HEADS) return;
  int h = t % HEADS;
  const float* row = H + (size_t)(t / HEADS) * F1 + h * HID;
  const float* vs = a_src + h * HID;
  const float* vd = a_dst + h * HID;
  float s = 0.f, d = 0.f;
#pragma unroll 8
  for (int c = 0; c < HID; ++c) { float x = row[c]; s += x * vs[c]; d += x * vd[c]; }
  as[t] = s; ad[t] = d;
}

__global__ void alphas2(const float* __restrict__ H2, const float* __restrict__ a_src,
                        const float* __restrict__ a_dst,
                        float* __restrict__ as, float* __restrict__ ad, int n) {
  int t = blockIdx.x * blockDim.x + threadIdx.x;
  if (t >= n) return;
  const float* row = H2 + (size_t)t * D_OUT;
  float s = 0.f, d = 0.f;
#pragma unroll
  for (int c = 0; c < D_OUT; ++c) { s += row[c] * a_src[c]; d += row[c] * a_dst[c]; }
  as[t] = s; ad[t] = d;
}

// =====================================================================
// segment-softmax passes (generic over head count Hn)
// =====================================================================
__device__ __forceinline__ void edge_nodes(int e, int nE, const int* src, const int* dst,
                                           int& s_, int& d_) {
  if (e < nE) { s_ = src[e]; d_ = dst[e]; }
  else        { s_ = d_ = e - nE; }         // self loops appended after edges
}

__global__ void edge_max(const int* __restrict__ src, const int* __restrict__ dst,
                         int nE, int n, const float* __restrict__ as,
                         const float* __restrict__ ad, unsigned* __restrict__ m, int Hn) {
  int t = blockIdx.x * blockDim.x + threadIdx.x;
  int total = (nE + n) * Hn;
  if (t >= total) return;
  int e = t / Hn, h = t - e * Hn;
  int s_, d_; edge_nodes(e, nE, src, dst, s_, d_);
  float ev = lrelu(as[(size_t)s_ * Hn + h] + ad[(size_t)d_ * Hn + h]);
  atomicMax(m + (size_t)d_ * Hn + h, f2ord(ev));
}

__global__ void edge_denom(const int* __restrict__ src, const int* __restrict__ dst,
                           int nE, int n, const float* __restrict__ as,
                           const float* __restrict__ ad, const unsigned* __restrict__ m,
                           float* __restrict__ denom, int Hn) {
  int t = blockIdx.x * blockDim.x + threadIdx.x;
  int total = (nE + n) * Hn;
  if (t >= total) return;
  int e = t / Hn, h = t - e * Hn;
  int s_, d_; edge_nodes(e, nE, src, dst, s_, d_);
  float ev = lrelu(as[(size_t)s_ * Hn + h] + ad[(size_t)d_ * Hn + h]);
  float w = __expf(ev - ord2f(m[(size_t)d_ * Hn + h]));
  atomicAdd(denom + (size_t)d_ * Hn + h, w);
}

// =====================================================================
// layer-1 message aggregation: one wave per edge, 16 floats per lane
// gather h1[src] (float4 loads), scale by per-head alpha, atomicAdd to out[dst]
// =====================================================================
__global__ void edge_agg1(const int* __restrict__ src, const int* __restrict__ dst,
                          int nE, int n, const float* __restrict__ H,
                          const float* __restrict__ as, const float* __restrict__ ad,
                          const unsigned* __restrict__ m, const float* __restrict__ denom,
                          float* __restrict__ out) {
  int e = (blockIdx.x * blockDim.x + threadIdx.x) >> 5;
  int lane = threadIdx.x & 31;
  if (e >= nE + n) return;
  int s_, d_; edge_nodes(e, nE, src, dst, s_, d_);
  int h = lane >> 2;                        // 16 contiguous cols per lane => fixed head
  float ev = lrelu(as[(size_t)s_ * HEADS + h] + ad[(size_t)d_ * HEADS + h]);
  float w = __expf(ev - ord2f(m[(size_t)d_ * HEADS + h]));
  float alpha = w / (denom[(size_t)d_ * HEADS + h] + 1e-16f);
  const float4* hp = (const float4*)(H + (size_t)s_ * F1 + lane * 16);
  float* op = out + (size_t)d_ * F1 + lane * 16;
#pragma unroll
  for (int j = 0; j < 4; ++j) {
    float4 v = hp[j];
    atomicAdd(op + j * 4 + 0, v.x * alpha);
    atomicAdd(op + j * 4 + 1, v.y * alpha);
    atomicAdd(op + j * 4 + 2, v.z * alpha);
    atomicAdd(op + j * 4 + 3, v.w * alpha);
  }
}

// x2 = relu(out1 + b1), in place
__global__ void relu_bias(float* __restrict__ X2, const float* __restrict__ b1, int total) {
  int t = blockIdx.x * blockDim.x + threadIdx.x;
  if (t >= total) return;
  float v = X2[t] + b1[t & (F1 - 1)];
  X2[t] = v > 0.f ? v : 0.f;
}

// layer-2 aggregation: thread per (edge, c)
__global__ void edge_agg2(const int* __restrict__ src, const int* __restrict__ dst,
                          int nE, int n, const float* __restrict__ H2,
                          const float* __restrict__ as, const float* __restrict__ ad,
                          const unsigned* __restrict__ m, const float* __restrict__ denom,
                          float* __restrict__ out) {
  int t = blockIdx.x * blockDim.x + threadIdx.x;
  int total = (nE + n) * D_OUT;
  if (t >= total) return;
  int e = t >> 3, c = t & 7;
  int s_, d_; edge_nodes(e, nE, src, dst, s_, d_);
  float ev = lrelu(as[s_] + ad[d_]);
  float alpha = __expf(ev - ord2f(m[d_])) / (denom[d_] + 1e-16f);
  atomicAdd(out + (size_t)d_ * D_OUT + c, H2[(size_t)s_ * D_OUT + c] * alpha);
}

__global__ void finalize(const float* __restrict__ acc, const float* __restrict__ b2,
                         float* __restrict__ out, int total) {
  int t = blockIdx.x * blockDim.x + threadIdx.x;
  if (t >= total) return;
  out[t] = acc[t] + b2[t & (D_OUT - 1)];
}

// =====================================================================
extern "C" void kernel_launch(void* const* d_in, const int* in_sizes, int n_in,
                              void* d_out, int out_size, void* d_ws, size_t ws_size,
                              hipStream_t stream) {
  const float* x      = (const float*)d_in[0];
  const int*   ei     = (const int*)  d_in[1];
  const float* W1     = (const float*)d_in[2];
  const float* a_src1 = (const float*)d_in[3];
  const float* a_dst1 = (const float*)d_in[4];
  const float* b1     = (const float*)d_in[5];
  const float* W2     = (const float*)d_in[6];
  const float* a_src2 = (const float*)d_in[7];
  const float* a_dst2 = (const float*)d_in[8];
  const float* b2     = (const float*)d_in[9];

  int n  = in_sizes[0] / D_IN;    // 50000
  int nE = in_sizes[1] / 2;       // 500000
  const int* src = ei;
  const int* dst = ei + nE;
  int EE = nE + n;

  // ---- workspace layout ----
  char* ws = (char*)d_ws;
  size_t off = 0;
  auto alloc = [&](size_t bytes) -> char* {
    char* p = ws + off; off += (bytes + 255) & ~(size_t)255; return p;
  };
  float*    h1   = (float*)   alloc((size_t)n * F1 * 4);
  float*    out1 = (float*)   alloc((size_t)n * F1 * 4);   // becomes x2 after relu_bias
  float*    as1  = (float*)   alloc((size_t)n * HEADS * 4);
  float*    ad1  = (float*)   alloc((size_t)n * HEADS * 4);
  unsigned* m1   = (unsigned*)alloc((size_t)n * HEADS * 4);
  float*    dn1  = (float*)   alloc((size_t)n * HEADS * 4);
  float*    h2   = (float*)   alloc((size_t)n * D_OUT * 4);
  float*    as2  = (float*)   alloc((size_t)n * 4);
  float*    ad2  = (float*)   alloc((size_t)n * 4);
  unsigned* m2   = (unsigned*)alloc((size_t)n * 4);
  float*    dn2  = (float*)   alloc((size_t)n * 4);
  float*    o2   = (float*)   alloc((size_t)n * D_OUT * 4);

  // zero all accumulators/state (everything past h1; h1 is fully overwritten)
  hipMemsetAsync(out1, 0, off - (size_t)((char*)out1 - ws), stream);

  // ---- layer 1 ----
  {
    int waves  = ((n + 15) / 16) * (F1 / 16);          // 100000
    int blocks = (waves + 7) / 8;
    gemm1_wmma<<<blocks, 256, 0, stream>>>(x, W1, h1, n);
  }
  alphas1<<<(n * HEADS + 255) / 256, 256, 0, stream>>>(h1, a_src1, a_dst1, as1, ad1, n);
  {
    int total = EE * HEADS;
    edge_max  <<<(total + 255) / 256, 256, 0, stream>>>(src, dst, nE, n, as1, ad1, m1, HEADS);
    edge_denom<<<(total + 255) / 256, 256, 0, stream>>>(src, dst, nE, n, as1, ad1, m1, dn1, HEADS);
  }
  edge_agg1<<<(EE + 7) / 8, 256, 0, stream>>>(src, dst, nE, n, h1, as1, ad1, m1, dn1, out1);
  relu_bias<<<((n * F1) + 255) / 256, 256, 0, stream>>>(out1, b1, n * F1);

  // ---- layer 2 ----
  {
    int waves  = (n + 15) / 16;
    int blocks = (waves + 7) / 8;
    gemm2_wmma<<<blocks, 256, 0, stream>>>(out1, W2, h2, n);
  }
  alphas2<<<(n + 255) / 256, 256, 0, stream>>>(h2, a_src2, a_dst2, as2, ad2, n);
  edge_max  <<<(EE + 255) / 256, 256, 0, stream>>>(src, dst, nE, n, as2, ad2, m2, 1);
  edge_denom<<<(EE + 255) / 256, 256, 0, stream>>>(src, dst, nE, n, as2, ad2, m2, dn2, 1);
  edge_agg2<<<(EE * D_OUT + 255) / 256, 256, 0, stream>>>(src, dst, nE, n, h2, as2, ad2, m2, dn2, o2);

  finalize<<<(n * D_OUT + 255) / 256, 256, 0, stream>>>(o2, b2, (float*)d_out, n * D_OUT);
}